// MFilmMambaDense_27822798144210
// MI455X (gfx1250) — compile-verified
//
#include <hip/hip_runtime.h>
#include <hip/hip_bf16.h>
#include <math.h>

// Problem constants (from reference)
#define B_SZ    64
#define L_SZ    88200
#define T_SZ    8820          // BLOCK_NUM
#define POOL    10            // L / BLOCK_NUM
#define NSTATE  16
#define OUTC    256           // 2 * OUT_DIM

typedef __attribute__((ext_vector_type(2))) float v2f;
typedef __attribute__((ext_vector_type(8))) float v8f;

__device__ __forceinline__ float silu_f(float x) {
    return x / (1.0f + __expf(-x));
}
__device__ __forceinline__ float softplus_f(float x) {
    return (x > 20.0f) ? x : log1pf(__expf(x));
}

// ---------------------------------------------------------------------------
// Kernel 1: max-pool x (B, L) -> pooled (B, T), window 10
// ---------------------------------------------------------------------------
__global__ void pool_kernel(const float* __restrict__ x, float* __restrict__ pooled) {
    int idx = blockIdx.x * blockDim.x + threadIdx.x;
    if (idx >= B_SZ * T_SZ) return;
    int b = idx / T_SZ;
    int t = idx - b * T_SZ;
    const float* p = x + (size_t)b * L_SZ + (size_t)t * POOL;
    float m = p[0];
#pragma unroll
    for (int i = 1; i < POOL; ++i) m = fmaxf(m, p[i]);
    pooled[idx] = m;
}

// ---------------------------------------------------------------------------
// Kernel 2: fully fused bidirectional Mamba scan.
// One wave32 per (batch, direction): lane = ch*16 + n  (ch in {0,1}, n state).
// All pre-scan math (in_proj, causal conv, silu, x_proj, softplus) is
// recomputed per step from the pooled scalar (D_MODEL == 1).
// Writes hcat (B, T, 2): [...,0] = hf, [...,1] = hb (already un-flipped).
// ---------------------------------------------------------------------------
__global__ __launch_bounds__(32) void bimamba_scan_kernel(
    const float* __restrict__ pooled,      // (B, T)
    const float* __restrict__ in_proj_w,   // (2,1,4)
    const float* __restrict__ conv_w,      // (2,2,4)
    const float* __restrict__ conv_b,      // (2,2)
    const float* __restrict__ x_proj_w,    // (2,2,33)
    const float* __restrict__ dt_proj_w,   // (2,1,2)
    const float* __restrict__ dt_proj_b,   // (2,2)
    const float* __restrict__ A_log,       // (2,2,16)
    const float* __restrict__ D_ssm,       // (2,2)
    const float* __restrict__ out_proj_w,  // (2,2,1)
    float* __restrict__ hcat)              // (B, T, 2)
{
    const int wid  = blockIdx.x;
    const int b    = wid >> 1;
    const int d    = wid & 1;
    const int lane = threadIdx.x;
    const int ch   = lane >> 4;
    const int n    = lane & 15;

    // direction-uniform scalars
    const float inw0  = in_proj_w[d * 4 + 0];
    const float inw1  = in_proj_w[d * 4 + 1];
    const float cw00 = conv_w[d * 8 + 0], cw01 = conv_w[d * 8 + 1];
    const float cw02 = conv_w[d * 8 + 2], cw03 = conv_w[d * 8 + 3];
    const float cw10 = conv_w[d * 8 + 4], cw11 = conv_w[d * 8 + 5];
    const float cw12 = conv_w[d * 8 + 6], cw13 = conv_w[d * 8 + 7];
    const float cb0  = conv_b[d * 2 + 0];
    const float cb1  = conv_b[d * 2 + 1];
    const float xw00 = x_proj_w[d * 66 + 0];       // dt col, row 0
    const float xw10 = x_proj_w[d * 66 + 33];      // dt col, row 1

    // per-lane constants
    const float dtw_ch = dt_proj_w[d * 2 + ch];
    const float dtb_ch = dt_proj_b[d * 2 + ch];
    const float A_n    = -__expf(A_log[d * 32 + ch * 16 + n]);
    const float Dp_ch  = D_ssm[d * 2 + ch];
    const float ow_ch  = out_proj_w[d * 2 + ch];
    const float zw_ch  = in_proj_w[d * 4 + 2 + ch];
    const float xwB0   = x_proj_w[d * 66 +      1 + n];   // B col n, row 0
    const float xwB1   = x_proj_w[d * 66 + 33 + 1 + n];   // B col n, row 1
    const float xwC0   = x_proj_w[d * 66 +      17 + n];  // C col n, row 0
    const float xwC1   = x_proj_w[d * 66 + 33 + 17 + n];  // C col n, row 1

    float hm3 = 0.0f, hm2 = 0.0f, hm1 = 0.0f;  // rolling conv window
    float hstate = 0.0f;                       // SSM state for (ch, n)

    const float* prow = pooled + (size_t)b * T_SZ;
    float* wrow = hcat + (size_t)b * T_SZ * 2 + d;

    for (int t = 0; t < T_SZ; ++t) {
        const int tt = d ? (T_SZ - 1 - t) : t;
        const float hv = prow[tt];

        // depthwise causal conv (w[3] multiplies x[t]) then silu
        const float c0v = cb0 + inw0 * (cw00 * hm3 + cw01 * hm2 + cw02 * hm1 + cw03 * hv);
        const float c1v = cb1 + inw1 * (cw10 * hm3 + cw11 * hm2 + cw12 * hm1 + cw13 * hv);
        const float xs0 = silu_f(c0v);
        const float xs1 = silu_f(c1v);

        const float dt_raw = xs0 * xw00 + xs1 * xw10;
        const float dt     = softplus_f(dt_raw * dtw_ch + dtb_ch);
        const float Bc     = xs0 * xwB0 + xs1 * xwB1;
        const float Cc     = xs0 * xwC0 + xs1 * xwC1;
        const float u      = (ch == 0) ? xs0 : xs1;

        const float dA = __expf(dt * A_n);
        hstate = dA * hstate + dt * Bc * u;

        // y[ch] = sum_n hstate * Cc  (reduce within each 16-lane half)
        float part = hstate * Cc;
        part += __shfl_xor(part, 1, 32);
        part += __shfl_xor(part, 2, 32);
        part += __shfl_xor(part, 4, 32);
        part += __shfl_xor(part, 8, 32);
        const float y = part + u * Dp_ch;

        const float z = hv * zw_ch;
        float g = y * silu_f(z) * ow_ch;
        g += __shfl_xor(g, 16, 32);     // combine the two channels

        if (lane == 0) wrow[(size_t)tt * 2] = hv + g;   // residual + mamba out

        hm3 = hm2; hm2 = hm1; hm1 = hv;
    }
}

// ---------------------------------------------------------------------------
// Kernel 3: FiLM MLP via V_WMMA_F32_16X16X4_F32, transposed GEMM.
//   D[c_tile, t_tile] (16x16) = mlp_w^T (16x2, K padded to 4) x hcat^T (2x16)
// M = output channels, N = time: per-VGPR stores hit 16 consecutive t and
// successive VGPRs are +T rows (24-bit immediate offsets). Bias is folded
// into the WMMA C operand. Operand masking is branchless (v_cndmask, no EXEC
// juggling); the gamma/beta split is straight-line SALU arithmetic; only the
// single tail t-tile takes the predicated path.
// Output: gamma (B,128,T) ++ beta (B,128,T), fully written.
// ---------------------------------------------------------------------------
__global__ __launch_bounds__(256) void film_mlp_wmma_kernel(
    const float* __restrict__ hcat,   // (B, T, 2)
    const float* __restrict__ mlp_w,  // (2, 256)
    const float* __restrict__ mlp_b,  // (256)
    float* __restrict__ out)          // gamma ++ beta
{
    const int lane  = threadIdx.x;           // 0..31
    const int wv    = threadIdx.y;           // 0..7
    const int b     = blockIdx.y;
    const int tTile = blockIdx.x * 8 + wv;   // 69*8 = 552 tiles = ceil(8820/16)
    const int t0    = tTile * 16;
    const int half  = lane >> 4;
    const int ln    = lane & 15;
    const int t     = t0 + ln;

    const size_t G    = (size_t)B_SZ * 128 * T_SZ;   // gamma half size
    const size_t bOff = (size_t)b * 128 * T_SZ;

    // B operand (4x16, K x N=t), branchless: all lanes load (upper half
    // re-reads the same cachelines), zeroed by select. Clamped index keeps
    // the tail tile in-bounds.
    const int tc = (t < T_SZ) ? t : (T_SZ - 1);
    const float* hp = hcat + ((size_t)b * T_SZ + tc) * 2;
    const float hb0 = hp[0];
    const float hb1 = hp[1];
    const bool bvalid = (half == 0) && (t < T_SZ);
    v2f bm;
    bm.x = bvalid ? hb0 : 0.0f;
    bm.y = bvalid ? hb1 : 0.0f;

    const bool fullTile = (t0 + 16 <= T_SZ);   // wave-uniform

    if (fullTile) {
#pragma unroll 8
        for (int j = 0; j < 16; ++j) {
            const int c0 = j * 16;

            // A operand (16x4, M=c x K): branchless zero of K=2,3 half.
            const float a0 = mlp_w[c0 + ln];
            const float a1 = mlp_w[256 + c0 + ln];
            v2f am;
            am.x = half ? 0.0f : a0;
            am.y = half ? 0.0f : a1;

            // Bias folded into C: rows c0 + half*8 + v are contiguous.
            v8f acc;
            const float* bp = mlp_b + c0 + half * 8;
#pragma unroll
            for (int v = 0; v < 8; ++v) acc[v] = bp[v];

            acc = __builtin_amdgcn_wmma_f32_16x16x4_f32(
                false, am, false, bm, (short)0, acc, false, false);

            // gamma/beta split without branches: (j>>3) picks the half,
            // (c0 & 127) is the channel row within it.
            float* p = out + (size_t)(j >> 3) * G + bOff
                           + (size_t)(c0 & 127) * T_SZ
                           + (size_t)(half * 8) * T_SZ + (size_t)t;
#pragma unroll
            for (int v = 0; v < 8; ++v) p[(size_t)v * T_SZ] = acc[v];
        }
    } else {
        const bool tok = (t < T_SZ);
#pragma unroll 4
        for (int j = 0; j < 16; ++j) {
            const int c0 = j * 16;

            const float a0 = mlp_w[c0 + ln];
            const float a1 = mlp_w[256 + c0 + ln];
            v2f am;
            am.x = half ? 0.0f : a0;
            am.y = half ? 0.0f : a1;

            v8f acc;
            const float* bp = mlp_b + c0 + half * 8;
#pragma unroll
            for (int v = 0; v < 8; ++v) acc[v] = bp[v];

            acc = __builtin_amdgcn_wmma_f32_16x16x4_f32(
                false, am, false, bm, (short)0, acc, false, false);

            float* p = out + (size_t)(j >> 3) * G + bOff
                           + (size_t)(c0 & 127) * T_SZ
                           + (size_t)(half * 8) * T_SZ + (size_t)t;
            if (tok) {
#pragma unroll
                for (int v = 0; v < 8; ++v) p[(size_t)v * T_SZ] = acc[v];
            }
        }
    }
}

// ---------------------------------------------------------------------------
extern "C" void kernel_launch(void* const* d_in, const int* in_sizes, int n_in,
                              void* d_out, int out_size, void* d_ws, size_t ws_size,
                              hipStream_t stream) {
    const float* x          = (const float*)d_in[0];
    const float* in_proj_w  = (const float*)d_in[1];
    const float* conv_w     = (const float*)d_in[2];
    const float* conv_b     = (const float*)d_in[3];
    const float* x_proj_w   = (const float*)d_in[4];
    const float* dt_proj_w  = (const float*)d_in[5];
    const float* dt_proj_b  = (const float*)d_in[6];
    const float* A_log      = (const float*)d_in[7];
    const float* D_ssm      = (const float*)d_in[8];
    const float* out_proj_w = (const float*)d_in[9];
    const float* mlp_w      = (const float*)d_in[10];
    const float* mlp_b      = (const float*)d_in[11];
    float* out = (float*)d_out;

    float* pooled = (float*)d_ws;                       // B*T floats (2.3 MB)
    float* hcat   = pooled + (size_t)B_SZ * T_SZ;       // B*T*2 floats (4.5 MB)

    {
        const int nTot = B_SZ * T_SZ;
        pool_kernel<<<(nTot + 255) / 256, 256, 0, stream>>>(x, pooled);
    }

    bimamba_scan_kernel<<<B_SZ * 2, 32, 0, stream>>>(
        pooled, in_proj_w, conv_w, conv_b, x_proj_w,
        dt_proj_w, dt_proj_b, A_log, D_ssm, out_proj_w, hcat);

    {
        dim3 grid(69, B_SZ);   // 69*8 waves = 552 t-tiles = ceil(8820/16)
        dim3 blk(32, 8);
        film_mlp_wmma_kernel<<<grid, blk, 0, stream>>>(hcat, mlp_w, mlp_b, out);
    }
}